// SecurePooling_33964601376866
// MI455X (gfx1250) — compile-verified
//
#include <hip/hip_runtime.h>

typedef __attribute__((ext_vector_type(2))) float v2f;
typedef __attribute__((ext_vector_type(8))) float v8f;

// Problem constants: x is (32, 1000, 14, 14) f32.  xs = {0,2,4,6,8}, K=6, M=5.
// All SAT coordinates needed are even -> only sat[2p][2q], p,q in 0..7 ("sat8").
#define NB   32
#define NC   1000
#define NCH  (NB*NC)
#define HW   196           // 14*14
#define SAT8 64            // 8x8 floats per channel
#define NTWO 625           // 5^4 (i,j,k,l)
#define NONE 25            // 5x5 (i,j)
#define NENT 651           // 625 two + 25 one + 1 total
#define NCHUNK 8
#define CPC  (NC/NCHUNK)   // 125 channels per chunk

// ---------------------------------------------------------------------------
// CDNA5 async global->LDS copies (ASYNCcnt path).  Flat pointers to LDS carry
// the LDS byte offset in bits [31:0] (ISA 10.2 aperture mapping), so a 32-bit
// truncation of the __shared__ pointer is the VDST LDS address.
// ---------------------------------------------------------------------------
__device__ __forceinline__ void async_ld_b128(const float* gp, const float* lp) {
    unsigned l = (unsigned)(size_t)lp;
    asm volatile("global_load_async_to_lds_b128 %0, %1, off"
                 :: "v"(l), "v"(gp) : "memory");
}
__device__ __forceinline__ void s_wait_async0() {
    asm volatile("s_wait_asynccnt 0x0" ::: "memory");
}
__device__ __forceinline__ void s_wait_async1() {
    asm volatile("s_wait_asynccnt 0x1" ::: "memory");
}

// ---------------------------------------------------------------------------
// Kernel 1: per-channel SAT via WMMA.  One wave (32 lanes) per channel.
//   T = L * X   (L = 16x16 lower-triangular ones; X = 14x14 tile, virtual pad)
//   S = T * L^T ->  S[i][j] = sum_{r<=i,c<=j} X[r][c] = sat[i+1][j+1]
//   sat8[p][q] = S[2p-1][2q-1]  (p,q >= 1), border row/col = 0.
// Image staged with 49 async b128 copies (784 B, 16B-aligned, no VGPR transit).
// Virtual zero-padding is done by CLAMPING the gather index to a zero sentinel
// slot (tr[196] == 0) -> unconditional ds_load + v_cndmask on the address,
// no EXEC-predicated load blocks.
// ---------------------------------------------------------------------------
__global__ void __launch_bounds__(256) k1_sat_wmma(const float* __restrict__ x,
                                                   float* __restrict__ satWS) {
    __shared__ float tR[8][200];   // raw 14x14 tile (196 floats) + zero sentinel
    __shared__ float tT[8][320];   // T staging, row stride 20 (conflict-free)

    const int wid  = threadIdx.x >> 5;
    const int lane = threadIdx.x & 31;
    const int ch   = blockIdx.x * 8 + wid;          // exactly 32000 channels
    const float* g = x + (size_t)ch * HW;
    float* tr = tR[wid];
    float* tt = tT[wid];

    // Zero sentinel (read by out-of-bounds clamped gathers).
    if (lane < 4) tr[196 + lane] = 0.0f;

    // Async bulk copy: 49 x b128 (lanes 0..16 issue twice, rest once).
    for (int q4 = lane; q4 < 49; q4 += 32)
        async_ld_b128(g + q4 * 4, &tr[q4 * 4]);
    s_wait_async0();

    const int hw   = lane >> 4;     // half-wave select
    const int m    = lane & 15;     // A-row index
    const int n    = m;             // B/C column index
    const bool ncol = (n < 14);     // per-lane column validity

    // ---- T = L * X ---------------------------------------------------------
    v8f acc = {};
#pragma unroll
    for (int k = 0; k < 16; k += 4) {
        const int r0 = k + 2 * hw;  // K = r0, r0+1 for this half-wave
        v2f a, b;
        a.x = (r0     <= m) ? 1.0f : 0.0f;                    // L[m][r0]
        a.y = (r0 + 1 <= m) ? 1.0f : 0.0f;
        // Clamped-index gather: OOB lanes read the zero sentinel tr[196].
        const int i0 = (r0     < 14 && ncol) ? ( r0      * 14 + n) : 196;
        const int i1 = (r0 + 1 < 14 && ncol) ? ((r0 + 1) * 14 + n) : 196;
        b.x = tr[i0];                                         // X[r0][n]   (or 0)
        b.y = tr[i1];                                         // X[r0+1][n] (or 0)
        acc = __builtin_amdgcn_wmma_f32_16x16x4_f32(false, a, false, b,
                                                    (short)0, acc, false, false);
    }

    // Re-layout T (C-layout -> row-major LDS) so it can feed the A operand.
    union { v8f v; float f[8]; } uT; uT.v = acc;
#pragma unroll
    for (int v = 0; v < 8; ++v) tt[(v + 8 * hw) * 20 + n] = uT.f[v];

    // ---- S = T * L^T -------------------------------------------------------
    v8f s = {};
#pragma unroll
    for (int k = 0; k < 16; k += 4) {
        const int c0 = k + 2 * hw;
        v2f a, b;
        a.x = tt[m * 20 + c0];                  // T[m][c0]
        a.y = tt[m * 20 + c0 + 1];
        b.x = (c0     <= n) ? 1.0f : 0.0f;      // L^T[c0][n] = (c0 <= n)
        b.y = (c0 + 1 <= n) ? 1.0f : 0.0f;
        s = __builtin_amdgcn_wmma_f32_16x16x4_f32(false, a, false, b,
                                                  (short)0, s, false, false);
    }

    // ---- Emit sat8 (8x8): border zeros + 49 interior values ---------------
    union { v8f v; float f[8]; } uS; uS.v = s;
    float* o = satWS + (size_t)ch * SAT8;
    if (lane < 8)       o[lane] = 0.0f;             // row p=0
    else if (lane < 15) o[(lane - 7) * 8] = 0.0f;   // col q=0, p=1..7
#pragma unroll
    for (int v = 0; v < 8; ++v) {
        const int M = v + 8 * hw;
        if ((M & 1) && (n & 1) && M < 14 && n < 14) {
            const int p = (M + 1) >> 1, q = (n + 1) >> 1;
            o[p * 8 + q] = uS.f[v];               // sat8[p][q] = S[2p-1][2q-1]
        }
    }
}

// ---------------------------------------------------------------------------
// Kernel 2: per-(b, channel-chunk) partial argmax over channels for all 651
// logit streams.  sat8 double-buffered in LDS via async b128 loads: iteration
// c issues the load for c+1, then s_wait_asynccnt 1 (in-order completion =>
// the older load for c has landed) before the visibility barrier.
// ---------------------------------------------------------------------------
__global__ void __launch_bounds__(672) k2_partial(const float* __restrict__ satWS,
                                                  float* __restrict__ pval,
                                                  int*   __restrict__ pidx) {
    __shared__ float s8[2][SAT8];
    const int b     = blockIdx.x >> 3;
    const int chunk = blockIdx.x & 7;
    const int t     = threadIdx.x;

    // Loop-invariant sat8 indices for this thread's logit stream.
    int a0 = 0, a1 = 0, a2 = 0, a3 = 0;           // winsum(i,j)
    int b0 = 0, b1 = 0, b2 = 0, b3 = 0;           // winsum(k,l)
    int o0 = 0, o1 = 0, o2 = 0, o3 = 0;           // overlap
    if (t < NTWO) {
        const int i = t / 125, j = (t / 25) % 5, k = (t / 5) % 5, l = t % 5;
        a0 = (i + 3) * 8 + (j + 3); a1 = i * 8 + (j + 3);
        a2 = (i + 3) * 8 + j;       a3 = i * 8 + j;
        b0 = (k + 3) * 8 + (l + 3); b1 = k * 8 + (l + 3);
        b2 = (k + 3) * 8 + l;       b3 = k * 8 + l;
        const int LOik = max(i, k), HIik = max(min(i, k) + 3, LOik);
        const int LOjl = max(j, l), HIjl = max(min(j, l) + 3, LOjl);
        o0 = HIik * 8 + HIjl; o1 = LOik * 8 + HIjl;
        o2 = HIik * 8 + LOjl; o3 = LOik * 8 + LOjl;
    } else if (t < NTWO + NONE) {
        const int e = t - NTWO, i = e / 5, j = e % 5;
        a0 = (i + 3) * 8 + (j + 3); a1 = i * 8 + (j + 3);
        a2 = (i + 3) * 8 + j;       a3 = i * 8 + j;
    }

    const int c0 = chunk * CPC, cend = c0 + CPC;
    if (t < 16)                                    // preload first channel
        async_ld_b128(satWS + (size_t)(b * NC + c0) * SAT8 + t * 4, &s8[0][t * 4]);

    float best = -3.4e38f;
    int   bidx = 0;
    for (int c = c0; c < cend; ++c) {
        const int cb = (c - c0) & 1;
        __syncthreads();                           // done reading buf[cb^1]
        const bool more = (c + 1) < cend;
        if (t < 16 && more)                        // prefetch next channel
            async_ld_b128(satWS + (size_t)(b * NC + c + 1) * SAT8 + t * 4,
                          &s8[cb ^ 1][t * 4]);
        if (more) s_wait_async1(); else s_wait_async0();
        __syncthreads();                           // buf[cb] visible to all

        const float* sb = s8[cb];
        const float tot = sb[63];
        float val = -3.4e38f;
        if (t < NTWO) {
            const float wij = sb[a0] - sb[a1] - sb[a2] + sb[a3];
            const float wkl = sb[b0] - sb[b1] - sb[b2] + sb[b3];
            const float ov  = sb[o0] - sb[o1] - sb[o2] + sb[o3];
            val = tot - wij - wkl + ov;
        } else if (t < NTWO + NONE) {
            const float wij = sb[a0] - sb[a1] - sb[a2] + sb[a3];
            val = tot - wij;
        } else if (t == NTWO + NONE) {
            val = tot;
        }
        if (val > best) { best = val; bidx = c; }  // strict > keeps first max
    }
    if (t < NENT) {
        pval[(size_t)blockIdx.x * NENT + t] = best;
        pidx[(size_t)blockIdx.x * NENT + t] = bidx;
    }
}

// ---------------------------------------------------------------------------
// Kernel 3: combine chunk partials (ascending chunk order preserves jnp's
// first-max tie rule) and run the disagree/agree/first-candidate logic.
// ---------------------------------------------------------------------------
__global__ void __launch_bounds__(672) k3_final(const float* __restrict__ pval,
                                                const int*   __restrict__ pidx,
                                                int* __restrict__ out) {
    __shared__ int twoI[NTWO];
    __shared__ int oneI[NONE];
    __shared__ int predI;
    __shared__ int cand[NONE];
    const int b = blockIdx.x;
    const int t = threadIdx.x;

    if (t < NENT) {
        float best = -3.4e38f;
        int   bi   = 0;
        for (int ch = 0; ch < NCHUNK; ++ch) {
            const size_t o = (size_t)(b * NCHUNK + ch) * NENT + t;
            const float v = pval[o];
            if (v > best) { best = v; bi = pidx[o]; }
        }
        if (t < NTWO)             twoI[t] = bi;
        else if (t < NTWO + NONE) oneI[t - NTWO] = bi;
        else                      predI = bi;
    }
    __syncthreads();
    if (t < NONE) {
        const int om = oneI[t];
        int ag = 1;
        for (int m = 0; m < NONE; ++m) ag &= (twoI[t * NONE + m] == om);
        cand[t] = ag && (om != predI);
    }
    __syncthreads();
    if (t == 0) {
        int res = predI;
        for (int q = NONE - 1; q >= 0; --q)       // descending scan -> first true wins
            if (cand[q]) res = oneI[q];
        out[b] = res;
    }
}

// ---------------------------------------------------------------------------
extern "C" void kernel_launch(void* const* d_in, const int* in_sizes, int n_in,
                              void* d_out, int out_size, void* d_ws, size_t ws_size,
                              hipStream_t stream) {
    (void)in_sizes; (void)n_in; (void)out_size; (void)ws_size;
    const float* x = (const float*)d_in[0];
    int* out = (int*)d_out;

    float* satWS = (float*)d_ws;                         // 32000*64 f32 = 8.19 MB
    float* pval  = satWS + (size_t)NCH * SAT8;           // 256*651 f32
    int*   pidx  = (int*)(pval + (size_t)NB * NCHUNK * NENT);

    k1_sat_wmma<<<NCH / 8, 256, 0, stream>>>(x, satWS);
    k2_partial<<<NB * NCHUNK, 672, 0, stream>>>(satWS, pval, pidx);
    k3_final<<<NB, 672, 0, stream>>>(pval, pidx, out);
}